// GraphSAGE_21157008900372
// MI455X (gfx1250) — compile-verified
//
#include <hip/hip_runtime.h>
#include <math.h>

typedef __attribute__((ext_vector_type(2))) float v2f;
typedef __attribute__((ext_vector_type(8))) float v8f;

#define NN 100000
#define EE 1600000

// Force the hardware GLOBAL_ATOMIC_ADD_F32 (no CAS loop, fire-and-forget).
__device__ __forceinline__ void atomic_add_f32(float* p, float v) {
    unsafeAtomicAdd(p, v);
}

// ---------------------------------------------------------------- zero (float4)
__global__ void zero_kernel(float4* __restrict__ p, long n4) {
    long i = (long)blockIdx.x * blockDim.x + threadIdx.x;
    long st = (long)gridDim.x * blockDim.x;
    for (; i < n4; i += st) p[i] = make_float4(0.f, 0.f, 0.f, 0.f);
}

// ---------------------------------------------------------------- degree count
__global__ void degree_kernel(const int* __restrict__ dst, float* __restrict__ deg) {
    int i = blockIdx.x * blockDim.x + threadIdx.x;
    int st = gridDim.x * blockDim.x;
    for (int e = i; e < EE; e += st)
        atomic_add_f32(&deg[dst[e]], 1.0f);
}

__global__ void inv_deg_kernel(float* __restrict__ deg) {
    int i = blockIdx.x * blockDim.x + threadIdx.x;
    int st = gridDim.x * blockDim.x;
    for (; i < NN; i += st)
        deg[i] = 1.0f / fmaxf(deg[i], 1.0f);
}

// ---------------------------------------------------------------- edge scatter-add
// agg[dst] += xin[src], 4 floats per thread via float4 gather + f32 atomics
template <int D>
__global__ void scatter_kernel(const float* __restrict__ xin,
                               const int* __restrict__ src,
                               const int* __restrict__ dst,
                               float* __restrict__ agg) {
    const int CH = D / 4;                       // float4 chunks per row
    long i = (long)blockIdx.x * blockDim.x + threadIdx.x;
    long st = (long)gridDim.x * blockDim.x;
    const long total = (long)EE * CH;
    for (long t = i; t < total; t += st) {
        int e = (int)(t / CH);
        int c = (int)(t % CH);
        int s = src[e];
        int d = dst[e];
        const float4 v = *(const float4*)(xin + (long)s * D + c * 4);
        float* o = agg + (long)d * D + c * 4;
        atomic_add_f32(o + 0, v.x);
        atomic_add_f32(o + 1, v.y);
        atomic_add_f32(o + 2, v.z);
        atomic_add_f32(o + 3, v.w);
    }
}

// ---------------------------------------------------------------- fused SAGE GEMM
// out[m, :] = act( (agg[m,:] * dinv[m]) @ Wl^T + xin[m,:] @ Wr^T + bias )
// One wave per 16-row block; NT 16-wide column tiles; WMMA f32 16x16x4.
// A layout: lane l -> row l%16, k = kstep + 2*(l/16) + {0,1}  (two VGPRs)
// B layout mirrors A with N = l%16. C/D: vgpr j -> M = j + 8*(l/16), N = l%16.
template <int K, int OUT, int NT, bool RELU>
__global__ __launch_bounds__(32) void sage_gemm_kernel(
        const float* __restrict__ agg,   // [N,K]
        const float* __restrict__ xin,   // [N,K]
        const float* __restrict__ Wl,    // [OUT,K] row-major
        const float* __restrict__ Wr,    // [OUT,K]
        const float* __restrict__ bias,  // [OUT]
        const float* __restrict__ dinv,  // [N]
        float* __restrict__ out) {       // [N,OUT]
    const int lane  = threadIdx.x;       // 0..31 (wave32)
    const int mlane = lane & 15;
    const int khalf = (lane >> 4) * 2;   // 0 or 2
    const int rowbase = blockIdx.x * 16;
    const int row = rowbase + mlane;
    const float s = dinv[row];

    v8f c[NT] = {};

    const float* arow = agg + (long)row * K;
    const float* xrow = xin + (long)row * K;

    // phase 1: mean-aggregated neighbours through Wl
    for (int kk = 0; kk < K; kk += 4) {
        v2f a = *(const v2f*)(arow + kk + khalf);
        a = a * s;
#pragma unroll
        for (int t = 0; t < NT; ++t) {
            const int col = t * 16 + mlane;
            v2f b;
            if ((OUT % 16 == 0) || (col < OUT))
                b = *(const v2f*)(Wl + (long)col * K + kk + khalf);
            else
                b = (v2f){0.f, 0.f};
            c[t] = __builtin_amdgcn_wmma_f32_16x16x4_f32(
                false, a, false, b, (short)0, c[t], false, false);
        }
    }
    // phase 2: self features through Wr
    for (int kk = 0; kk < K; kk += 4) {
        v2f a = *(const v2f*)(xrow + kk + khalf);
#pragma unroll
        for (int t = 0; t < NT; ++t) {
            const int col = t * 16 + mlane;
            v2f b;
            if ((OUT % 16 == 0) || (col < OUT))
                b = *(const v2f*)(Wr + (long)col * K + kk + khalf);
            else
                b = (v2f){0.f, 0.f};
            c[t] = __builtin_amdgcn_wmma_f32_16x16x4_f32(
                false, a, false, b, (short)0, c[t], false, false);
        }
    }
    // epilogue: bias (+ReLU) and store
#pragma unroll
    for (int t = 0; t < NT; ++t) {
        const int col = t * 16 + mlane;
        if ((OUT % 16 == 0) || (col < OUT)) {
            const float bb = bias[col];
#pragma unroll
            for (int j = 0; j < 8; ++j) {
                const int m = j + (lane >> 4) * 8;
                float v = c[t][j] + bb;
                if (RELU) v = fmaxf(v, 0.f);
                out[(long)(rowbase + m) * OUT + col] = v;
            }
        }
    }
}

// ---------------------------------------------------------------- log_softmax (40 cols)
__global__ void logsoftmax_kernel(float* __restrict__ out) {
    const int wave = threadIdx.x >> 5;
    const int lane = threadIdx.x & 31;
    const int row = blockIdx.x * (blockDim.x >> 5) + wave;
    if (row >= NN) return;
    float* p = out + (long)row * 40;
    float v0 = (lane < 40) ? p[lane] : -1e30f;
    float v1 = (lane + 32 < 40) ? p[lane + 32] : -1e30f;
    float m = fmaxf(v0, v1);
    for (int off = 16; off > 0; off >>= 1) m = fmaxf(m, __shfl_xor(m, off, 32));
    float e = ((lane < 40) ? expf(v0 - m) : 0.f) +
              ((lane + 32 < 40) ? expf(v1 - m) : 0.f);
    for (int off = 16; off > 0; off >>= 1) e += __shfl_xor(e, off, 32);
    const float lse = logf(e) + m;
    if (lane < 40) p[lane] = v0 - lse;
    if (lane + 32 < 40) p[lane + 32] = v1 - lse;
}

// ---------------------------------------------------------------- host launcher
extern "C" void kernel_launch(void* const* d_in, const int* in_sizes, int n_in,
                              void* d_out, int out_size, void* d_ws, size_t ws_size,
                              hipStream_t stream) {
    const float* x   = (const float*)d_in[0];
    const int*  edge = (const int*)d_in[1];     // [2,E] int32
    const float* Wl0 = (const float*)d_in[2];
    const float* Wr0 = (const float*)d_in[3];
    const float* b0  = (const float*)d_in[4];
    const float* Wl1 = (const float*)d_in[5];
    const float* Wr1 = (const float*)d_in[6];
    const float* b1  = (const float*)d_in[7];
    const float* Wlo = (const float*)d_in[8];
    const float* Wro = (const float*)d_in[9];
    const float* bo  = (const float*)d_in[10];
    const int* src = edge;
    const int* dst = edge + EE;

    // workspace: deg[N] | bufA[N*256] | bufB[N*256]   (~205.2 MB)
    float* deg  = (float*)d_ws;
    float* bufA = deg + NN;                    // 400000 B offset, 16B aligned
    float* bufB = bufA + (size_t)NN * 256;
    float* outp = (float*)d_out;

    // degree -> 1/max(deg,1)
    zero_kernel<<<1024, 256, 0, stream>>>((float4*)deg, NN / 4);
    degree_kernel<<<4096, 256, 0, stream>>>(dst, deg);
    inv_deg_kernel<<<512, 256, 0, stream>>>(deg);

    // ---- layer 0: x(128) -> h0(256) in bufB
    zero_kernel<<<8192, 256, 0, stream>>>((float4*)bufA, (long)NN * 128 / 4);
    scatter_kernel<128><<<16384, 256, 0, stream>>>(x, src, dst, bufA);
    sage_gemm_kernel<128, 256, 16, true><<<NN / 16, 32, 0, stream>>>(
        bufA, x, Wl0, Wr0, b0, deg, bufB);

    // ---- layer 1: h0(256) -> h1(256), in-place in bufB (wave owns its rows)
    zero_kernel<<<8192, 256, 0, stream>>>((float4*)bufA, (long)NN * 256 / 4);
    scatter_kernel<256><<<16384, 256, 0, stream>>>(bufB, src, dst, bufA);
    sage_gemm_kernel<256, 256, 16, true><<<NN / 16, 32, 0, stream>>>(
        bufA, bufB, Wl1, Wr1, b1, deg, bufB);

    // ---- output layer: h1(256) -> out(40)
    zero_kernel<<<8192, 256, 0, stream>>>((float4*)bufA, (long)NN * 256 / 4);
    scatter_kernel<256><<<16384, 256, 0, stream>>>(bufB, src, dst, bufA);
    sage_gemm_kernel<256, 40, 3, false><<<NN / 16, 32, 0, stream>>>(
        bufA, bufB, Wlo, Wro, bo, deg, outp);

    // ---- log_softmax over 40 classes, in place on d_out
    logsoftmax_kernel<<<(NN + 7) / 8, 256, 0, stream>>>(outp);
}